// STEM_89919435309224
// MI455X (gfx1250) — compile-verified
//
#include <hip/hip_runtime.h>
#include <hip/hip_bf16.h>

// ---- problem constants (match reference) ----
#define N_PTS   8192
#define KNN     32
#define N_EDGE  (N_PTS * KNN)          // 262144
#define RADIUS  (0.02f * 2.1f)
#define NEGBIG  (-1e30f)
#define BN_EPS  (1e-5f)
#define BIGD    (1e30f)
#define LRELU(v) ((v) > 0.0f ? (v) : 0.01f * (v))

typedef __attribute__((ext_vector_type(2))) float v2f;
typedef __attribute__((ext_vector_type(8))) float v8f;

// ------------------------------------------------------------------
// 0) zero the BN accumulators (graph-capture safe; runs every call)
// ------------------------------------------------------------------
__global__ void init_ws_kernel(unsigned int* cnt, float* gsum, float* gsumsq) {
    int t = threadIdx.x;
    if (t == 0) *cnt = 0u;
    if (t < 64) { gsum[t] = 0.0f; gsumsq[t] = 0.0f; }
}

// ------------------------------------------------------------------
// 1) pos4 = [pos / sf[batch], reflectance]; sq = |pos|^2
// ------------------------------------------------------------------
__global__ void __launch_bounds__(256) pos4_kernel(
    const float* __restrict__ pos, const float* __restrict__ refl,
    const float* __restrict__ sf, const int* __restrict__ batch,
    float* __restrict__ pos4, float* __restrict__ sq) {
    int i = blockIdx.x * blockDim.x + threadIdx.x;
    if (i >= N_PTS) return;
    float x = pos[i * 3 + 0], y = pos[i * 3 + 1], z = pos[i * 3 + 2];
    sq[i] = x * x + y * y + z * z;
    float inv = 1.0f / sf[batch[i]];
    pos4[i * 4 + 0] = x * inv;
    pos4[i * 4 + 1] = y * inv;
    pos4[i * 4 + 2] = z * inv;
    pos4[i * 4 + 3] = refl[i];
}

// ------------------------------------------------------------------
// 2) brute-force radius search, K nearest kept (LDS-tiled candidates)
//    one thread per query; 32 tiles of 256 candidates staged in LDS
// ------------------------------------------------------------------
__global__ void __launch_bounds__(256) topk_kernel(
    const float* __restrict__ pos, const float* __restrict__ sq,
    const int* __restrict__ batch,
    int* __restrict__ nbr, int* __restrict__ maskArr,
    unsigned int* __restrict__ cnt) {
    __shared__ float spx[256], spy[256], spz[256], ssq[256];
    __shared__ int   sbt[256];

    const int tid = threadIdx.x;
    const int i   = blockIdx.x * 256 + tid;

    const float qx = pos[i * 3 + 0], qy = pos[i * 3 + 1], qz = pos[i * 3 + 2];
    const float qsq = sq[i];
    const int   qb  = batch[i];
    const float r2  = RADIUS * RADIUS;

    float bd[KNN];
    int   bi[KNN];
    for (int k = 0; k < KNN; ++k) { bd[k] = BIGD; bi[k] = i; }

    for (int t = 0; t < N_PTS / 256; ++t) {
        int j = t * 256 + tid;
        spx[tid] = pos[j * 3 + 0];
        spy[tid] = pos[j * 3 + 1];
        spz[tid] = pos[j * 3 + 2];
        ssq[tid] = sq[j];
        sbt[tid] = batch[j];
        if (t + 1 < N_PTS / 256)  // hint next tile into cache
            __builtin_prefetch(&pos[(t + 1) * 256 * 3 + tid * 3], 0, 0);
        __syncthreads();
        for (int u = 0; u < 256; ++u) {
            float d2 = qsq + ssq[u] - 2.0f * (qx * spx[u] + qy * spy[u] + qz * spz[u]);
            if (d2 < r2 && sbt[u] == qb && d2 < bd[KNN - 1]) {
                int p = KNN - 1;
                while (p > 0 && bd[p - 1] > d2) {
                    bd[p] = bd[p - 1]; bi[p] = bi[p - 1]; --p;
                }
                bd[p] = d2; bi[p] = t * 256 + u;
            }
        }
        __syncthreads();
    }

    int nv = 0;
    for (int k = 0; k < KNN; ++k) {
        int v = (bd[k] < BIGD * 0.5f) ? 1 : 0;
        maskArr[i * KNN + k] = v;
        nbr[i * KNN + k]     = bi[k];
        nv += v;
    }
    atomicAdd(cnt, (unsigned int)nv);
}

// ------------------------------------------------------------------
// 3) fused: gather 8-d edge features -> layer1 (8->64, LDS) ->
//    layer2 (64->64) via V_WMMA_F32_16X16X4_F32 -> h2 + BN partials
//    block = 128 threads (4 waves); tile = 16 edges x 64 channels
//    wave w owns channels [16w, 16w+16)
// ------------------------------------------------------------------
__global__ void __launch_bounds__(128) mlp_wmma_kernel(
    const float* __restrict__ xin, const float* __restrict__ pos4,
    const int* __restrict__ nbr, const int* __restrict__ maskArr,
    const float* __restrict__ w1, const float* __restrict__ b1,
    const float* __restrict__ w2, const float* __restrict__ b2,
    float* __restrict__ h2, float* __restrict__ gsum, float* __restrict__ gsumsq) {
    __shared__ float feats[16 * 8];      // 16 edges x 8 features
    __shared__ float w1s[8 * 64];        // layer1 weights
    __shared__ float h1s[16 * 68];       // 16 edges x 64 (stride 68: bank-safe)
    __shared__ float lsum[64], lsumsq[64];
    __shared__ int   smask[16];

    const int t  = threadIdx.x;
    const int eb = blockIdx.x * 16;      // edge tile base

    // stage w1 + edge features + mask
    for (int u = t; u < 512; u += 128) w1s[u] = w1[u];
    {
        int e = eb + (t >> 3);
        int f = t & 7;
        int j = nbr[e];
        int i = e >> 5;                  // e / KNN
        float v = (f < 4) ? xin[j * 4 + f]
                          : (pos4[j * 4 + (f - 4)] - pos4[i * 4 + (f - 4)]);
        feats[t] = v;
    }
    if (t < 16) smask[t] = maskArr[eb + t];
    if (t < 64) { lsum[t] = 0.0f; lsumsq[t] = 0.0f; }
    __syncthreads();

    // layer1: each thread -> 1 edge row m, 8 channels
    {
        int m  = t >> 3;
        int cb = (t & 7) * 8;
        #pragma unroll
        for (int cc = 0; cc < 8; ++cc) {
            int c = cb + cc;
            float acc = b1[c];
            #pragma unroll
            for (int d = 0; d < 8; ++d)
                acc += feats[m * 8 + d] * w1s[d * 64 + c];
            h1s[m * 68 + c] = LRELU(acc);
        }
    }
    __syncthreads();

    // layer2: 16x16 output tile per wave, K=64 as 16 steps of f32 WMMA K=4
    const int lane = t & 31;
    const int wv   = t >> 5;
    const int n0   = wv * 16;
    const int m16  = lane & 15;          // A: row M; B: col N
    const int khi  = lane >> 4;          // lanes 16-31 carry K+2,K+3
    const int c    = n0 + m16;           // output channel for this lane

    v8f acc = {0.f, 0.f, 0.f, 0.f, 0.f, 0.f, 0.f, 0.f};
    #pragma unroll
    for (int kk = 0; kk < 16; ++kk) {
        int k0 = kk * 4 + 2 * khi;
        v2f a, b;
        a.x = h1s[m16 * 68 + k0];
        a.y = h1s[m16 * 68 + k0 + 1];
        b.x = w2[k0 * 64 + c];
        b.y = w2[(k0 + 1) * 64 + c];
        acc = __builtin_amdgcn_wmma_f32_16x16x4_f32(
            /*neg_a=*/false, a, /*neg_b=*/false, b,
            /*c_mod=*/(short)0, acc, /*reuse_a=*/false, /*reuse_b=*/false);
    }

    const float bias = b2[c];
    float ls = 0.0f, lss = 0.0f;
    #pragma unroll
    for (int r = 0; r < 8; ++r) {
        int M = r + 8 * khi;             // C/D layout: VGPR r, lane half -> row
        float v = acc[r] + bias;
        v = LRELU(v);
        h2[(size_t)(eb + M) * 64 + c] = v;
        if (smask[M]) { ls += v; lss += v * v; }
    }
    atomicAdd(&lsum[c], ls);
    atomicAdd(&lsumsq[c], lss);
    __syncthreads();
    if (t < 64) {
        atomicAdd(&gsum[t],   lsum[t]);
        atomicAdd(&gsumsq[t], lsumsq[t]);
    }
}

// ------------------------------------------------------------------
// 4) BN finalize: scale = gamma/sqrt(var+eps), shift = beta - mean*scale
//    var = E[h^2] - mean^2 over masked edges (== reference's masked var)
// ------------------------------------------------------------------
__global__ void bn_finalize_kernel(
    const float* __restrict__ gsum, const float* __restrict__ gsumsq,
    const unsigned int* __restrict__ cnt,
    const float* __restrict__ gamma, const float* __restrict__ beta,
    float* __restrict__ scale, float* __restrict__ shift) {
    int c = threadIdx.x;
    if (c >= 64) return;
    float n    = (float)(*cnt);
    float mean = gsum[c] / n;
    float var  = gsumsq[c] / n - mean * mean;
    float s    = gamma[c] * rsqrtf(var + BN_EPS);
    scale[c] = s;
    shift[c] = beta[c] - mean * s;
}

// ------------------------------------------------------------------
// 5) masked max aggregation over K neighbors
// ------------------------------------------------------------------
__global__ void __launch_bounds__(256) aggregate_kernel(
    const float* __restrict__ h2, const int* __restrict__ maskArr,
    const float* __restrict__ scale, const float* __restrict__ shift,
    float* __restrict__ out) {
    int g = blockIdx.x * 256 + threadIdx.x;     // over N*64
    int i = g >> 6;
    int c = g & 63;
    float s = scale[c], sh = shift[c];
    float best = NEGBIG;
    for (int k = 0; k < KNN; ++k) {
        int e = i * KNN + k;
        if (maskArr[e]) {
            float v = h2[(size_t)e * 64 + c] * s + sh;
            best = v > best ? v : best;
        }
    }
    out[g] = best;
}

// ------------------------------------------------------------------
// 6) passthrough outputs: pos, batch, reflectance, sf
// ------------------------------------------------------------------
__global__ void __launch_bounds__(256) tail_kernel(
    const float* __restrict__ pos, const int* __restrict__ batch,
    const float* __restrict__ refl, const float* __restrict__ sf,
    float* __restrict__ out) {
    int i = blockIdx.x * 256 + threadIdx.x;
    if (i < N_PTS) {
        out[N_PTS * 64 + i * 3 + 0] = pos[i * 3 + 0];
        out[N_PTS * 64 + i * 3 + 1] = pos[i * 3 + 1];
        out[N_PTS * 64 + i * 3 + 2] = pos[i * 3 + 2];
        out[N_PTS * 67 + i] = (float)batch[i];
        out[N_PTS * 68 + i] = refl[i];
    }
    if (i < 4) out[N_PTS * 69 + i] = sf[i];
}

// ------------------------------------------------------------------
extern "C" void kernel_launch(void* const* d_in, const int* in_sizes, int n_in,
                              void* d_out, int out_size, void* d_ws, size_t ws_size,
                              hipStream_t stream) {
    const float* x      = (const float*)d_in[0];
    const float* pos    = (const float*)d_in[1];
    const float* refl   = (const float*)d_in[2];
    const float* sf     = (const float*)d_in[3];
    const float* w1     = (const float*)d_in[4];
    const float* b1     = (const float*)d_in[5];
    const float* w2     = (const float*)d_in[6];
    const float* b2     = (const float*)d_in[7];
    const float* gamma  = (const float*)d_in[8];
    const float* beta   = (const float*)d_in[9];
    const int*   batch  = (const int*)d_in[10];
    float*       out    = (float*)d_out;

    // workspace layout (all offsets 256B-aligned)
    char* p = (char*)d_ws;
    unsigned int* cnt   = (unsigned int*)(p + 0);
    float* gsum         = (float*)(p + 256);
    float* gsumsq       = (float*)(p + 512);
    float* scale        = (float*)(p + 768);
    float* shift        = (float*)(p + 1024);
    float* pos4         = (float*)(p + 4096);                         // N*4 f32
    float* sq           = (float*)(p + 4096 + N_PTS * 16);            // N f32
    int*   nbr          = (int*)  (p + 4096 + N_PTS * 20);            // E i32
    int*   maskArr      = (int*)  (p + 4096 + N_PTS * 20 + N_EDGE * 4);
    float* h2           = (float*)(p + 4096 + N_PTS * 20 + N_EDGE * 8); // E*64 f32

    init_ws_kernel<<<1, 64, 0, stream>>>(cnt, gsum, gsumsq);
    pos4_kernel<<<N_PTS / 256, 256, 0, stream>>>(pos, refl, sf, batch, pos4, sq);
    topk_kernel<<<N_PTS / 256, 256, 0, stream>>>(pos, sq, batch, nbr, maskArr, cnt);
    mlp_wmma_kernel<<<N_EDGE / 16, 128, 0, stream>>>(x, pos4, nbr, maskArr,
                                                     w1, b1, w2, b2,
                                                     h2, gsum, gsumsq);
    bn_finalize_kernel<<<1, 64, 0, stream>>>(gsum, gsumsq, cnt, gamma, beta, scale, shift);
    aggregate_kernel<<<(N_PTS * 64) / 256, 256, 0, stream>>>(h2, maskArr, scale, shift, out);
    tail_kernel<<<N_PTS / 256, 256, 0, stream>>>(pos, batch, refl, sf, out);
}